// Idealizer_19602230739704
// MI455X (gfx1250) — compile-verified
//
#include <hip/hip_runtime.h>
#include <math.h>
#include <stdint.h>

// ---------- gfx1250 async global->LDS staging (ASYNCcnt path, ISA 15.18.3) ----------
__device__ __forceinline__ void async_copy_b32(uint32_t lds_addr, const void* gaddr) {
    asm volatile("global_load_async_to_lds_b32 %0, %1, off"
                 :
                 : "v"(lds_addr), "v"((uint64_t)(uintptr_t)gaddr)
                 : "memory");
}
__device__ __forceinline__ void async_copy_b128(uint32_t lds_addr, const void* gaddr) {
    asm volatile("global_load_async_to_lds_b128 %0, %1, off"
                 :
                 : "v"(lds_addr), "v"((uint64_t)(uintptr_t)gaddr)
                 : "memory");
}
__device__ __forceinline__ void wait_asynccnt0() {
    asm volatile("s_wait_asynccnt 0x0" ::: "memory");
}
__device__ __forceinline__ uint32_t lds_addr32(const void* p) {
    // generic LDS address = {shared aperture, lds_offset[31:0]} -> low 32 bits are the DS address
    return (uint32_t)(uintptr_t)p;
}

// ---------- small fixed-size linear algebra (row-major 3x3) ----------
__device__ __forceinline__ void cross3(const float a[3], const float b[3], float o[3]) {
    o[0] = a[1] * b[2] - a[2] * b[1];
    o[1] = a[2] * b[0] - a[0] * b[2];
    o[2] = a[0] * b[1] - a[1] * b[0];
}
__device__ __forceinline__ float dot3(const float a[3], const float b[3]) {
    return a[0] * b[0] + a[1] * b[1] + a[2] * b[2];
}
__device__ __forceinline__ void mm3(const float a[9], const float b[9], float c[9]) {
#pragma unroll
    for (int i = 0; i < 3; ++i)
#pragma unroll
        for (int j = 0; j < 3; ++j)
            c[3 * i + j] = a[3 * i + 0] * b[0 + j] + a[3 * i + 1] * b[3 + j] + a[3 * i + 2] * b[6 + j];
}
__device__ __forceinline__ void mv3(const float a[9], const float v[3], float o[3]) {
#pragma unroll
    for (int i = 0; i < 3; ++i)
        o[i] = a[3 * i + 0] * v[0] + a[3 * i + 1] * v[1] + a[3 * i + 2] * v[2];
}

// dihedral with jnp.nan_to_num(atan2(...)) semantics (NaN -> 0)
__device__ __forceinline__ float dihedral(const float p0[3], const float p1[3],
                                          const float p2[3], const float p3[3]) {
    float b0[3], b1[3], b2[3];
#pragma unroll
    for (int k = 0; k < 3; ++k) {
        b0[k] = p0[k] - p1[k];   // -(p1 - p0)
        b1[k] = p2[k] - p1[k];
        b2[k] = p3[k] - p2[k];
    }
    float c1[3], c2[3], cc[3];
    cross3(b0, b1, c1);          // b0 x b1
    cross3(b2, b1, c2);          // b2 x b1
    cross3(c1, c2, cc);
    float y = dot3(cc, b1) / sqrtf(dot3(b1, b1));
    float x = dot3(c1, c2);
    float r = atan2f(y, x);
    return (r == r) ? r : 0.0f;  // nan_to_num
}

#define N_AA   21
#define N_FR   8
#define N_AT   14
#define LDS_DF (N_AA * N_FR * 16)     // 2688 floats: full 4x4 default frames (16B-multiple)
#define LDS_LP (N_AA * N_AT * 3)      // 882 floats
#define LDS_SC (N_AA * N_AT)          // 294

__global__ __launch_bounds__(256)
void idealizer_kernel(const int* __restrict__ aa,
                      const float* __restrict__ bb,        // (n,4,3)
                      const float* __restrict__ tor,       // (n,4)
                      const float* __restrict__ dfr,       // (21,8,4,4)
                      const int* __restrict__ gidx,        // (21,14)
                      const float* __restrict__ amask,     // (21,14)
                      const float* __restrict__ lit,       // (21,14,3)
                      float* __restrict__ out,             // (n,14,3)
                      int n)
{
    __shared__ __align__(16) float s_df[LDS_DF];
    __shared__ float s_lit[LDS_LP];
    __shared__ float s_am[LDS_SC];
    __shared__ int   s_gi[LDS_SC];

    // ---- async-DMA the amino-acid tables into LDS (no VGPR round-trip) ----
    for (int t = threadIdx.x; t < LDS_DF / 4; t += blockDim.x)
        async_copy_b128(lds_addr32(&s_df[4 * t]), dfr + 4 * t);
    for (int t = threadIdx.x; t < LDS_LP; t += blockDim.x)
        async_copy_b32(lds_addr32(&s_lit[t]), lit + t);
    for (int t = threadIdx.x; t < LDS_SC; t += blockDim.x) {
        async_copy_b32(lds_addr32(&s_am[t]), amask + t);
        async_copy_b32(lds_addr32(&s_gi[t]), gidx + t);
    }

    const int i = blockIdx.x * blockDim.x + threadIdx.x;
    const float* row = bb + (size_t)i * 12;
    // gfx1250 prefetch of the streamed neighbor row (speculative; OOB is dropped)
    __builtin_prefetch(row + 12, 0, 0);

    wait_asynccnt0();      // our own async copies have landed in LDS
    __syncthreads();       // everyone's copies visible block-wide

    if (i >= n) return;

    float Nv[3], CA[3], Cv[3], O4[3];
#pragma unroll
    for (int k = 0; k < 3; ++k) {
        Nv[k] = row[0 + k];
        CA[k] = row[3 + k];
        Cv[k] = row[6 + k];
        O4[k] = row[9 + k];
    }
    // neighbor atoms (C of i-1; N, CA of i+1)
    float Cm[3] = {0.f, 0.f, 0.f}, Np[3] = {0.f, 0.f, 0.f}, CAp[3] = {0.f, 0.f, 0.f};
    if (i > 0) {
        const float* rm = bb + (size_t)(i - 1) * 12;
#pragma unroll
        for (int k = 0; k < 3; ++k) Cm[k] = rm[6 + k];
    }
    if (i < n - 1) {
        const float* rp = bb + (size_t)(i + 1) * 12;
#pragma unroll
        for (int k = 0; k < 3; ++k) { Np[k] = rp[0 + k]; CAp[k] = rp[3 + k]; }
    }

    // angles: [omega, phi, psi, chi1..chi4]
    float ang[7];
    ang[0] = (i < n - 1) ? dihedral(CA, Cv, Np, CAp) : 0.0f;   // omega
    ang[1] = (i > 0)     ? dihedral(Cm, Nv, CA, Cv)  : 0.0f;   // phi
    ang[2] = (i < n - 1) ? dihedral(Nv, CA, Cv, Np)  : 0.0f;   // psi
#pragma unroll
    for (int k = 0; k < 4; ++k) ang[3 + k] = tor[(size_t)i * 4 + k];

    float sv[8], cv[8];
    sv[0] = 0.0f; cv[0] = 1.0f;
#pragma unroll
    for (int k = 0; k < 7; ++k) sincosf(ang[k], &sv[k + 1], &cv[k + 1]);

    // ---------- backbone frame from N, CA, C ----------
    const float eps = 1e-20f;
    float nv[3], cvec[3];
#pragma unroll
    for (int k = 0; k < 3; ++k) { nv[k] = Nv[k] - CA[k]; cvec[k] = Cv[k] - CA[k]; }
    const float cx = cvec[0], cy = cvec[1], cz = cvec[2];
    float nrm  = sqrtf(eps + cx * cx + cy * cy);
    float s1 = -cy / nrm, c1 = cx / nrm;
    float nrm2 = sqrtf(eps + cx * cx + cy * cy + cz * cz);
    float s2 = cz / nrm2, c2 = sqrtf(cx * cx + cy * cy) / nrm2;
    // Rc = R2 @ R1
    float Rc[9] = { c2 * c1, -c2 * s1, s2,
                    s1,       c1,      0.0f,
                   -s2 * c1,  s2 * s1, c2 };
    float n2[3];
    mv3(Rc, nv, n2);
    float nrm3 = sqrtf(eps + n2[1] * n2[1] + n2[2] * n2[2]);
    float sn = -n2[2] / nrm3, cn = n2[1] / nrm3;
    // M = Rn @ Rc ; bb_r = M^T
    float M[9];
#pragma unroll
    for (int j = 0; j < 3; ++j) {
        M[0 + j] = Rc[0 + j];
        M[3 + j] = cn * Rc[3 + j] - sn * Rc[6 + j];
        M[6 + j] = sn * Rc[3 + j] + cn * Rc[6 + j];
    }
    float bbR[9];
#pragma unroll
    for (int r = 0; r < 3; ++r)
#pragma unroll
        for (int c = 0; c < 3; ++c) bbR[3 * r + c] = M[3 * c + r];
    const float* bbT = CA;

    // ---------- per-amino-acid tables from LDS ----------
    const int a21 = aa[i];
    const float* df = s_df + a21 * (N_FR * 16);
    int   g[10];
    float lv[30];
#pragma unroll
    for (int a = 0; a < 10; ++a) {
        const int at = a21 * N_AT + 4 + a;
        g[a] = s_gi[at];
#pragma unroll
        for (int k = 0; k < 3; ++k) lv[3 * a + k] = s_lit[3 * at + k];
    }
    float p[30];
#pragma unroll
    for (int k = 0; k < 30; ++k) p[k] = 0.0f;

    // ---------- frame walk: default frame * Rx(angle), chained for f>=5 ----------
    float chR[9], chT[3];          // chained transform (frame4 composed onward)
#pragma unroll
    for (int f = 0; f < N_FR; ++f) {
        const float* d = df + f * 16;
        const float s = sv[f], c = cv[f];
        float frm[9], ftv[3];
#pragma unroll
        for (int r = 0; r < 3; ++r) {
            frm[3 * r + 0] =  d[4 * r + 0];
            frm[3 * r + 1] =  d[4 * r + 1] * c + d[4 * r + 2] * s;
            frm[3 * r + 2] = -d[4 * r + 1] * s + d[4 * r + 2] * c;
            ftv[r]         =  d[4 * r + 3];
        }
        float curR[9], curT[3];
        if (f <= 4) {
#pragma unroll
            for (int k = 0; k < 9; ++k) curR[k] = frm[k];
#pragma unroll
            for (int k = 0; k < 3; ++k) curT[k] = ftv[k];
            if (f == 4) {
#pragma unroll
                for (int k = 0; k < 9; ++k) chR[k] = curR[k];
#pragma unroll
                for (int k = 0; k < 3; ++k) chT[k] = curT[k];
            }
        } else {
            float t0[3];
            mm3(chR, frm, curR);
            mv3(chR, ftv, t0);
#pragma unroll
            for (int k = 0; k < 3; ++k) curT[k] = t0[k] + chT[k];
#pragma unroll
            for (int k = 0; k < 9; ++k) chR[k] = curR[k];
#pragma unroll
            for (int k = 0; k < 3; ++k) chT[k] = curT[k];
        }
        // global frame = bb ∘ cur
        float gR[9], gT[3], t1[3];
        mm3(bbR, curR, gR);
        mv3(bbR, curT, t1);
#pragma unroll
        for (int k = 0; k < 3; ++k) gT[k] = t1[k] + bbT[k];

        // predicated accumulation of side-chain atoms mapped to this frame
#pragma unroll
        for (int a = 0; a < 10; ++a) {
            const float m = (g[a] == f) ? 1.0f : 0.0f;
            float v[3];
            mv3(gR, &lv[3 * a], v);
#pragma unroll
            for (int k = 0; k < 3; ++k) p[3 * a + k] = fmaf(m, v[k] + gT[k], p[3 * a + k]);
        }
    }

    // ---------- write output: bb_coords (4 atoms) ++ masked predictions (10 atoms) ----------
    float* o = out + (size_t)i * 42;
#pragma unroll
    for (int k = 0; k < 3; ++k) {
        o[0 + k] = Nv[k];
        o[3 + k] = CA[k];
        o[6 + k] = Cv[k];
        o[9 + k] = O4[k];
    }
#pragma unroll
    for (int a = 0; a < 10; ++a) {
        const float m = s_am[a21 * N_AT + 4 + a];
#pragma unroll
        for (int k = 0; k < 3; ++k) o[12 + 3 * a + k] = p[3 * a + k] * m;
    }
}

extern "C" void kernel_launch(void* const* d_in, const int* in_sizes, int n_in,
                              void* d_out, int out_size, void* d_ws, size_t ws_size,
                              hipStream_t stream) {
    const int*   aa  = (const int*)d_in[0];
    const float* bb  = (const float*)d_in[1];
    const float* tor = (const float*)d_in[2];
    const float* dfr = (const float*)d_in[3];
    const int*   gi  = (const int*)d_in[4];
    const float* am  = (const float*)d_in[5];
    const float* lit = (const float*)d_in[6];
    float* out = (float*)d_out;

    const int n = in_sizes[0];
    const int block = 256;                      // 8 wave32 per block
    const int grid = (n + block - 1) / block;
    hipLaunchKernelGGL(idealizer_kernel, dim3(grid), dim3(block), 0, stream,
                       aa, bb, tor, dfr, gi, am, lit, out, n);
}